// GAT_67765993996293
// MI455X (gfx1250) — compile-verified
//
#include <hip/hip_runtime.h>
#include <hip/hip_bf16.h>

// GAT: 3 layers, N=100000 nodes, E=1600000 edges (+N self loops), fp32.
// WMMA f32 16x16x4 with LDS-staged A tiles for the dense transforms;
// vectorized (b128) edge kernels with atomic segment softmax / scatter-add;
// fused BN+skip+ELU elementwise passes.

#define NN      100000
#define EE      1600000
#define IN_DIM  128
#define HIDC    64
#define NHEADS  4
#define HDIM    256      // NHEADS * HIDC
#define OUT_DIM 2
#define NEG_SLOPE 0.2f
#define BN_EPS    1e-5f
#define TPB 256
#define KPAD 4           // LDS row padding (floats): bank = 4*m mod 64, conflict-free

typedef __attribute__((ext_vector_type(2))) float v2f;
typedef __attribute__((ext_vector_type(8))) float v8f;

static inline int cdiv(long long a, long long b) { return (int)((a + b - 1) / b); }

// ---- order-preserving float <-> uint key for atomic float max ----
__device__ __forceinline__ unsigned fkey(float f) {
    unsigned u = __float_as_uint(f);
    return (u & 0x80000000u) ? ~u : (u | 0x80000000u);
}
__device__ __forceinline__ float funkey(unsigned k) {
    return (k & 0x80000000u) ? __uint_as_float(k & 0x7fffffffu)
                             : __uint_as_float(~k);
}
__device__ __forceinline__ float lrelu(float v) { return v > 0.f ? v : NEG_SLOPE * v; }
__device__ __forceinline__ float eluf(float v)  { return v > 0.f ? v : expf(v) - 1.0f; }

// ---------------- init ----------------
__global__ void k_fill_f32(float* p, float v, int n) {
    int i = blockIdx.x * blockDim.x + threadIdx.x;
    if (i < n) p[i] = v;
}
__global__ void k_fill_u32(unsigned* p, unsigned v, int n) {
    int i = blockIdx.x * blockDim.x + threadIdx.x;
    if (i < n) p[i] = v;
}

// ---------------- WMMA GEMM: C[M,256] = A[M,K] @ B[K,256] (+bias) ----------------
// block = 4 waves. A tile (16 x K) staged once per block into padded LDS with
// coalesced float4 loads; all 4 waves share it. Wave w computes cols [64w,64w+64).
// A-frag (16x4 f32): lane l -> m=l&15, k-half=(l>>4)*2, 2 consecutive K values.
// B-frag (4x16 f32): lane l -> n=l&15, same k-half striping (coalesced b32 x2).
__global__ __launch_bounds__(128)
void k_gemm_wmma(const float* __restrict__ A, const float* __restrict__ B,
                 const float* __restrict__ bias, float* __restrict__ C, int K) {
    __shared__ float As[16 * (256 + KPAD)];
    const int KP   = K + KPAD;
    const int t    = threadIdx.x;
    const int row0 = blockIdx.x * 16;

    // cooperative A-tile load: 16*K floats, float4 per thread per step
    const int nf4 = (16 * K) >> 2;                 // 512 (K=128) or 1024 (K=256)
    for (int i4 = t; i4 < nf4; i4 += 128) {
        const int idx = i4 << 2;
        const int r = idx / K, c = idx % K;
        const float4 v = *(const float4*)(A + (size_t)(row0 + r) * K + c);
        *(float4*)(&As[r * KP + c]) = v;           // 16B-aligned: KP*4 % 16 == 0
    }
    __syncthreads();

    const int wave = t >> 5;
    const int lane = t & 31;
    const int colbase = wave * 64;
    const int m  = lane & 15;
    const int kh = (lane >> 4) << 1;

    v8f acc[4] = {};
    for (int k = 0; k < K; k += 4) {
        v2f a;
        const float* ap = &As[m * KP + k + kh];    // ds_load_b64, conflict-free
        a.x = ap[0];
        a.y = ap[1];
#pragma unroll
        for (int j = 0; j < 4; ++j) {
            const int col = colbase + j * 16 + m;
            v2f b;
            b.x = B[(size_t)(k + kh)     * HDIM + col];
            b.y = B[(size_t)(k + kh + 1) * HDIM + col];
            acc[j] = __builtin_amdgcn_wmma_f32_16x16x4_f32(
                false, a, false, b, (short)0, acc[j], false, false);
        }
    }
    // C/D layout: VGPR i -> row i (lanes 0-15) / row 8+i (lanes 16-31)
    const int rbase = row0 + ((lane >> 4) ? 8 : 0);
#pragma unroll
    for (int j = 0; j < 4; ++j) {
        const int col = colbase + j * 16 + m;
        const float bb = bias ? bias[col] : 0.0f;
#pragma unroll
        for (int i = 0; i < 8; ++i)
            C[(size_t)(rbase + i) * HDIM + col] = acc[j][i] + bb;
    }
}

// ---------------- tiny GEMM for layer 2 (256 -> 2) ----------------
__global__ void k_gemm_small(const float* __restrict__ A, const float* __restrict__ W,
                             float* __restrict__ C, int K, int OC) {
    int n = blockIdx.x * blockDim.x + threadIdx.x;
    if (n >= NN) return;
    for (int o = 0; o < OC; ++o) {
        float s = 0.f;
        for (int k = 0; k < K; ++k) s += A[(size_t)n * K + k] * W[k * OC + o];
        C[(size_t)n * OC + o] = s;
    }
}

// ---------------- attention coefficients: asrc/adst[n,h] = <h[n,h,:], a[h,:]> ----------------
__global__ void k_alpha(const float* __restrict__ h, const float* __restrict__ aw_s,
                        const float* __restrict__ aw_d, float* __restrict__ asrc,
                        float* __restrict__ adst, int heads, int ch) {
    int i = blockIdx.x * blockDim.x + threadIdx.x;
    if (i >= NN * heads) return;
    int n = i / heads, hh = i % heads;
    const float* hp = h + (size_t)n * heads * ch + hh * ch;
    const float* ws = aw_s + hh * ch;
    const float* wd = aw_d + hh * ch;
    float s1 = 0.f, s2 = 0.f;
    for (int c = 0; c < ch; ++c) { float v = hp[c]; s1 += v * ws[c]; s2 += v * wd[c]; }
    asrc[i] = s1;
    adst[i] = s2;
}

// ================= H=4 specialized edge kernels (vectorized, 1 thread/edge) =================
__global__ void k_edge_max4(const long long* __restrict__ ei, const float4* __restrict__ asrc4,
                            const float4* __restrict__ adst4, float4* __restrict__ ebuf4,
                            unsigned* __restrict__ mkey) {
    int e = blockIdx.x * blockDim.x + threadIdx.x;
    if (e >= EE) return;
    int s = (int)ei[e], d = (int)ei[EE + e];
    float4 a = asrc4[s], b = adst4[d];
    float4 v;
    v.x = lrelu(a.x + b.x); v.y = lrelu(a.y + b.y);
    v.z = lrelu(a.z + b.z); v.w = lrelu(a.w + b.w);
    ebuf4[e] = v;
    unsigned* mk = mkey + d * 4;
    atomicMax(mk + 0, fkey(v.x)); atomicMax(mk + 1, fkey(v.y));
    atomicMax(mk + 2, fkey(v.z)); atomicMax(mk + 3, fkey(v.w));
}
__global__ void k_self_max4(const float4* __restrict__ asrc4, const float4* __restrict__ adst4,
                            uint4* __restrict__ mkey4) {
    int i = blockIdx.x * blockDim.x + threadIdx.x;
    if (i >= NN) return;
    float4 a = asrc4[i], b = adst4[i];
    uint4 mk = mkey4[i];
    unsigned k;
    k = fkey(lrelu(a.x + b.x)); if (k > mk.x) mk.x = k;
    k = fkey(lrelu(a.y + b.y)); if (k > mk.y) mk.y = k;
    k = fkey(lrelu(a.z + b.z)); if (k > mk.z) mk.z = k;
    k = fkey(lrelu(a.w + b.w)); if (k > mk.w) mk.w = k;
    mkey4[i] = mk;   // serialized after edge pass; plain max
}
__global__ void k_edge_exp4(const long long* __restrict__ ei, float4* __restrict__ ebuf4,
                            const uint4* __restrict__ mkey4, float* __restrict__ ssum) {
    int e = blockIdx.x * blockDim.x + threadIdx.x;
    if (e >= EE) return;
    int d = (int)ei[EE + e];
    float4 v = ebuf4[e];
    uint4 mk = mkey4[d];
    float4 w;
    w.x = expf(v.x - funkey(mk.x)); w.y = expf(v.y - funkey(mk.y));
    w.z = expf(v.z - funkey(mk.z)); w.w = expf(v.w - funkey(mk.w));
    ebuf4[e] = w;
    float* sp = ssum + d * 4;
    atomicAdd(sp + 0, w.x); atomicAdd(sp + 1, w.y);
    atomicAdd(sp + 2, w.z); atomicAdd(sp + 3, w.w);
}
__global__ void k_self_sum4(const float4* __restrict__ asrc4, const float4* __restrict__ adst4,
                            const uint4* __restrict__ mkey4, float4* __restrict__ ssum4) {
    int i = blockIdx.x * blockDim.x + threadIdx.x;
    if (i >= NN) return;
    float4 a = asrc4[i], b = adst4[i];
    uint4 mk = mkey4[i];
    float4 s = ssum4[i];
    s.x += expf(lrelu(a.x + b.x) - funkey(mk.x));
    s.y += expf(lrelu(a.y + b.y) - funkey(mk.y));
    s.z += expf(lrelu(a.z + b.z) - funkey(mk.z));
    s.w += expf(lrelu(a.w + b.w) - funkey(mk.w));
    ssum4[i] = s;
}
// block-per-edge weighted scatter-add: 64 threads, LDS-broadcast indices+alphas,
// coalesced float4 gather of h[src] (1KB row), 256 atomic f32 adds into agg[dst]
__global__ __launch_bounds__(64)
void k_edge_agg256(const long long* __restrict__ ei, const float* __restrict__ ebuf,
                   const float* __restrict__ ssum, const float* __restrict__ h,
                   float* __restrict__ agg) {
    __shared__ int   sh_sd[2];
    __shared__ float sh_alpha[NHEADS];
    const int e = blockIdx.x;
    const int t = threadIdx.x;
    if (t == 0) { sh_sd[0] = (int)ei[e]; sh_sd[1] = (int)ei[EE + e]; }
    __syncthreads();
    const int s = sh_sd[0], d = sh_sd[1];
    if (t < NHEADS)
        sh_alpha[t] = ebuf[(size_t)e * NHEADS + t] / (ssum[d * NHEADS + t] + 1e-16f);
    __syncthreads();
    const int c = t * 4;
    const float alpha = sh_alpha[c >> 6];
    const float4 hv = *(const float4*)(h + (size_t)s * HDIM + c);
    float* ap = agg + (size_t)d * HDIM + c;
    atomicAdd(ap + 0, hv.x * alpha);
    atomicAdd(ap + 1, hv.y * alpha);
    atomicAdd(ap + 2, hv.z * alpha);
    atomicAdd(ap + 3, hv.w * alpha);
}
__global__ void k_self_agg256(const float* __restrict__ asrc, const float* __restrict__ adst,
                              const unsigned* __restrict__ mkey, const float* __restrict__ ssum,
                              const float* __restrict__ h, float* __restrict__ agg) {
    int idx = blockIdx.x * blockDim.x + threadIdx.x;     // over NN * 64
    if (idx >= NN * (HDIM / 4)) return;
    const int n = idx >> 6;
    const int c = (idx & 63) * 4;
    const int hh = c >> 6;
    const int nh = n * NHEADS + hh;
    float v = lrelu(asrc[nh] + adst[nh]);
    float alpha = expf(v - funkey(mkey[nh])) / (ssum[nh] + 1e-16f);
    const size_t off = (size_t)n * HDIM + c;
    float4 hv = *(const float4*)(h + off);
    float4 av = *(float4*)(agg + off);
    av.x += hv.x * alpha; av.y += hv.y * alpha;
    av.z += hv.z * alpha; av.w += hv.w * alpha;
    *(float4*)(agg + off) = av;
}

// ================= generic (scalar) edge kernels — layer 2, H=1 =================
__global__ void k_edge_max(const long long* __restrict__ ei, const float* __restrict__ asrc,
                           const float* __restrict__ adst, float* __restrict__ ebuf,
                           unsigned* __restrict__ mkey, int heads) {
    long long idx = (long long)blockIdx.x * blockDim.x + threadIdx.x;
    if (idx >= (long long)EE * heads) return;
    int e = (int)(idx / heads), hh = (int)(idx % heads);
    int s = (int)ei[e], d = (int)ei[EE + e];
    float v = lrelu(asrc[s * heads + hh] + adst[d * heads + hh]);
    ebuf[idx] = v;
    atomicMax(&mkey[d * heads + hh], fkey(v));
}
__global__ void k_self_max(const float* __restrict__ asrc, const float* __restrict__ adst,
                           unsigned* __restrict__ mkey, int heads) {
    int i = blockIdx.x * blockDim.x + threadIdx.x;
    if (i >= NN * heads) return;
    unsigned k = fkey(lrelu(asrc[i] + adst[i]));
    if (k > mkey[i]) mkey[i] = k;
}
__global__ void k_edge_exp(const long long* __restrict__ ei, float* __restrict__ ebuf,
                           const unsigned* __restrict__ mkey, float* __restrict__ ssum,
                           int heads) {
    long long idx = (long long)blockIdx.x * blockDim.x + threadIdx.x;
    if (idx >= (long long)EE * heads) return;
    int e = (int)(idx / heads), hh = (int)(idx % heads);
    int d = (int)ei[EE + e];
    float w = expf(ebuf[idx] - funkey(mkey[d * heads + hh]));
    ebuf[idx] = w;
    atomicAdd(&ssum[d * heads + hh], w);
}
__global__ void k_self_sum(const float* __restrict__ asrc, const float* __restrict__ adst,
                           const unsigned* __restrict__ mkey, float* __restrict__ ssum,
                           int heads) {
    int i = blockIdx.x * blockDim.x + threadIdx.x;
    if (i >= NN * heads) return;
    float v = lrelu(asrc[i] + adst[i]);
    ssum[i] += expf(v - funkey(mkey[i]));
}
__global__ void k_edge_agg(const long long* __restrict__ ei, const float* __restrict__ ebuf,
                           const float* __restrict__ ssum, const float* __restrict__ h,
                           float* __restrict__ agg, int heads, int ch) {
    const int tot = heads * ch;
    long long idx = (long long)blockIdx.x * blockDim.x + threadIdx.x;
    if (idx >= (long long)EE * tot) return;
    int e = (int)(idx / tot), r = (int)(idx % tot), hh = r / ch;
    int s = (int)ei[e], d = (int)ei[EE + e];
    float alpha = ebuf[(size_t)e * heads + hh] / (ssum[d * heads + hh] + 1e-16f);
    atomicAdd(&agg[(size_t)d * tot + r], h[(size_t)s * tot + r] * alpha);
}
__global__ void k_self_agg(const float* __restrict__ asrc, const float* __restrict__ adst,
                           const unsigned* __restrict__ mkey, const float* __restrict__ ssum,
                           const float* __restrict__ h, float* __restrict__ agg,
                           int heads, int ch) {
    const int tot = heads * ch;
    long long idx = (long long)blockIdx.x * blockDim.x + threadIdx.x;
    if (idx >= (long long)NN * tot) return;
    int n = (int)(idx / tot), r = (int)(idx % tot), hh = r / ch;
    float v = lrelu(asrc[n * heads + hh] + adst[n * heads + hh]);
    float alpha = expf(v - funkey(mkey[n * heads + hh])) / (ssum[n * heads + hh] + 1e-16f);
    agg[idx] += h[idx] * alpha;
}

// ---------------- batchnorm (training-style: biased var over N) ----------------
__global__ __launch_bounds__(64)
void k_bn_stats256(const float* __restrict__ x, const float* __restrict__ bias,
                   float* __restrict__ sums, float* __restrict__ sumsq, int rows_per_blk) {
    const int c = threadIdx.x * 4;
    int r0 = blockIdx.x * rows_per_blk;
    int r1 = r0 + rows_per_blk; if (r1 > NN) r1 = NN;
    const float4 b = *(const float4*)(bias + c);
    float4 s = {0, 0, 0, 0}, q = {0, 0, 0, 0};
    for (int r = r0; r < r1; ++r) {
        float4 v = *(const float4*)(x + (size_t)r * HDIM + c);
        v.x += b.x; v.y += b.y; v.z += b.z; v.w += b.w;
        s.x += v.x; s.y += v.y; s.z += v.z; s.w += v.w;
        q.x += v.x * v.x; q.y += v.y * v.y; q.z += v.z * v.z; q.w += v.w * v.w;
    }
    atomicAdd(&sums[c + 0], s.x); atomicAdd(&sums[c + 1], s.y);
    atomicAdd(&sums[c + 2], s.z); atomicAdd(&sums[c + 3], s.w);
    atomicAdd(&sumsq[c + 0], q.x); atomicAdd(&sumsq[c + 1], q.y);
    atomicAdd(&sumsq[c + 2], q.z); atomicAdd(&sumsq[c + 3], q.w);
}
__global__ void k_bn_stats(const float* __restrict__ x, const float* __restrict__ bias,
                           float* __restrict__ sums, float* __restrict__ sumsq,
                           int C, int rows_per_blk) {
    int c = threadIdx.x;
    int r0 = blockIdx.x * rows_per_blk;
    int r1 = r0 + rows_per_blk; if (r1 > NN) r1 = NN;
    float s = 0.f, q = 0.f;
    float b = bias[c];
    for (int r = r0; r < r1; ++r) {
        float v = x[(size_t)r * C + c] + b;
        s += v; q += v * v;
    }
    atomicAdd(&sums[c], s);
    atomicAdd(&sumsq[c], q);
}
__global__ void k_bn_fin(const float* __restrict__ sums, const float* __restrict__ sumsq,
                         float* __restrict__ mu, float* __restrict__ rstd, int C) {
    int c = threadIdx.x;
    if (c >= C) return;
    float m = sums[c] / (float)NN;
    float v = sumsq[c] / (float)NN - m * m;
    mu[c] = m;
    rstd[c] = rsqrtf(v + BN_EPS);
}

// ---------------- fused bias + BN + skip (+residual out) + ELU, float4 ----------------
__global__ void k_apply_layer4(const float* __restrict__ agg, const float* __restrict__ bias,
                               const float* __restrict__ mu, const float* __restrict__ rstd,
                               const float* __restrict__ g, const float* __restrict__ be,
                               const float* __restrict__ skip, float* res_out,
                               float* __restrict__ act_out, int write_res) {
    int idx = blockIdx.x * blockDim.x + threadIdx.x;     // over NN * 64
    if (idx >= NN * (HDIM / 4)) return;
    const int c = (idx & 63) * 4;
    const size_t off = (size_t)idx * 4;
    float4 v  = *(const float4*)(agg + off);
    float4 b4 = *(const float4*)(bias + c);
    float4 m4 = *(const float4*)(mu + c);
    float4 r4 = *(const float4*)(rstd + c);
    float4 g4 = *(const float4*)(g + c);
    float4 e4 = *(const float4*)(be + c);
    float4 sk = *(const float4*)(skip + off);
    float4 y;
    y.x = g4.x * (v.x + b4.x - m4.x) * r4.x + e4.x + sk.x;
    y.y = g4.y * (v.y + b4.y - m4.y) * r4.y + e4.y + sk.y;
    y.z = g4.z * (v.z + b4.z - m4.z) * r4.z + e4.z + sk.z;
    y.w = g4.w * (v.w + b4.w - m4.w) * r4.w + e4.w + sk.w;
    if (write_res) *(float4*)(res_out + off) = y;
    float4 a;
    a.x = eluf(y.x); a.y = eluf(y.y); a.z = eluf(y.z); a.w = eluf(y.w);
    *(float4*)(act_out + off) = a;
}
__global__ void k_apply_out(const float* __restrict__ agg, const float* __restrict__ bias,
                            const float* __restrict__ mu, const float* __restrict__ rstd,
                            const float* __restrict__ g, const float* __restrict__ be,
                            float* __restrict__ out) {
    int i = blockIdx.x * blockDim.x + threadIdx.x;
    if (i >= NN * OUT_DIM) return;
    int c = i % OUT_DIM;
    float v = agg[i] + bias[c];
    out[i] = g[c] * (v - mu[c]) * rstd[c] + be[c];
}

extern "C" void kernel_launch(void* const* d_in, const int* in_sizes, int n_in,
                              void* d_out, int out_size, void* d_ws, size_t ws_size,
                              hipStream_t stream) {
    const float*     x  = (const float*)d_in[0];
    const long long* ei = (const long long*)d_in[1];   // int64 [2,E]
    const float *W0 = (const float*)d_in[2],  *aS0 = (const float*)d_in[3],
                *aD0 = (const float*)d_in[4], *b0  = (const float*)d_in[5];
    const float *W1 = (const float*)d_in[6],  *aS1 = (const float*)d_in[7],
                *aD1 = (const float*)d_in[8], *b1  = (const float*)d_in[9];
    const float *W2 = (const float*)d_in[10], *aS2 = (const float*)d_in[11],
                *aD2 = (const float*)d_in[12],*b2  = (const float*)d_in[13];
    const float *skipW = (const float*)d_in[14], *skipb = (const float*)d_in[15];
    const float *g0 = (const float*)d_in[16], *be0 = (const float*)d_in[17];
    const float *g1 = (const float*)d_in[18], *be1 = (const float*)d_in[19];
    const float *g2 = (const float*)d_in[20], *be2 = (const float*)d_in[21];
    float* out = (float*)d_out;

    // ---- workspace carve-up (floats; all offsets 16B-aligned) ----
    float* ws = (float*)d_ws;
    size_t o = 0;
    float*    f_h    = ws + o; o += (size_t)NN * HDIM;
    float*    f_agg  = ws + o; o += (size_t)NN * HDIM;
    float*    f_res  = ws + o; o += (size_t)NN * HDIM;
    float*    f_act  = ws + o; o += (size_t)NN * HDIM;
    float*    f_asrc = ws + o; o += (size_t)NN * NHEADS;
    float*    f_adst = ws + o; o += (size_t)NN * NHEADS;
    unsigned* f_m    = (unsigned*)(ws + o); o += (size_t)NN * NHEADS;
    float*    f_s    = ws + o; o += (size_t)NN * NHEADS;
    float*    f_e    = ws + o; o += (size_t)EE * NHEADS;
    float*    f_sum  = ws + o; o += HDIM;
    float*    f_sq   = ws + o; o += HDIM;
    float*    f_mu   = ws + o; o += HDIM;
    float*    f_rs   = ws + o; o += HDIM;

    const int gemm_blocks = NN / 16;   // 6250, exact
    const int gNH  = cdiv((long long)NN * NHEADS, TPB);
    const int gNHD = cdiv((long long)NN * HDIM, TPB);
    const int gE   = cdiv(EE, TPB);
    const int gN   = cdiv(NN, TPB);
    const int gN64 = cdiv((long long)NN * (HDIM / 4), TPB);

    for (int layer = 0; layer < 2; ++layer) {
        const float* inp = layer == 0 ? x : f_act;
        const float* W   = layer == 0 ? W0 : W1;
        const float* aS  = layer == 0 ? aS0 : aS1;
        const float* aD  = layer == 0 ? aD0 : aD1;
        const float* bb  = layer == 0 ? b0 : b1;
        const float* gg  = layer == 0 ? g0 : g1;
        const float* bee = layer == 0 ? be0 : be1;
        const int K      = layer == 0 ? IN_DIM : HDIM;

        k_gemm_wmma<<<gemm_blocks, 128, 0, stream>>>(inp, W, nullptr, f_h, K);
        if (layer == 0)  // projected skip: residual = x @ skipW + skipb
            k_gemm_wmma<<<gemm_blocks, 128, 0, stream>>>(x, skipW, skipb, f_res, IN_DIM);
        k_alpha<<<gNH, TPB, 0, stream>>>(f_h, aS, aD, f_asrc, f_adst, NHEADS, HIDC);
        k_fill_u32<<<gNH, TPB, 0, stream>>>(f_m, 0u, NN * NHEADS);
        k_fill_f32<<<gNH, TPB, 0, stream>>>(f_s, 0.f, NN * NHEADS);
        k_fill_f32<<<gNHD, TPB, 0, stream>>>(f_agg, 0.f, NN * HDIM);
        k_edge_max4<<<gE, TPB, 0, stream>>>(ei, (const float4*)f_asrc, (const float4*)f_adst,
                                            (float4*)f_e, f_m);
        k_self_max4<<<gN, TPB, 0, stream>>>((const float4*)f_asrc, (const float4*)f_adst,
                                            (uint4*)f_m);
        k_edge_exp4<<<gE, TPB, 0, stream>>>(ei, (float4*)f_e, (const uint4*)f_m, f_s);
        k_self_sum4<<<gN, TPB, 0, stream>>>((const float4*)f_asrc, (const float4*)f_adst,
                                            (const uint4*)f_m, (float4*)f_s);
        k_edge_agg256<<<EE, 64, 0, stream>>>(ei, f_e, f_s, f_h, f_agg);
        k_self_agg256<<<gN64, TPB, 0, stream>>>(f_asrc, f_adst, f_m, f_s, f_h, f_agg);
        k_fill_f32<<<cdiv(2 * HDIM, TPB), TPB, 0, stream>>>(f_sum, 0.f, 2 * HDIM);
        k_bn_stats256<<<cdiv(NN, 128), 64, 0, stream>>>(f_agg, bb, f_sum, f_sq, 128);
        k_bn_fin<<<1, HDIM, 0, stream>>>(f_sum, f_sq, f_mu, f_rs, HDIM);
        // layer0: skip=f_res (projected), keep residual; layer1: skip=f_res (identity)
        k_apply_layer4<<<gN64, TPB, 0, stream>>>(f_agg, bb, f_mu, f_rs, gg, bee,
                                                 f_res, f_res, f_act, layer == 0 ? 1 : 0);
    }

    // ================= layer 2 (256 -> 1x2, concat=False) =================
    k_gemm_small<<<gN, TPB, 0, stream>>>(f_act, W2, f_h, HDIM, OUT_DIM);
    k_alpha<<<gN, TPB, 0, stream>>>(f_h, aS2, aD2, f_asrc, f_adst, 1, OUT_DIM);
    k_fill_u32<<<gN, TPB, 0, stream>>>(f_m, 0u, NN);
    k_fill_f32<<<gN, TPB, 0, stream>>>(f_s, 0.f, NN);
    k_fill_f32<<<cdiv((long long)NN * OUT_DIM, TPB), TPB, 0, stream>>>(f_agg, 0.f, NN * OUT_DIM);
    k_edge_max<<<gE, TPB, 0, stream>>>(ei, f_asrc, f_adst, f_e, f_m, 1);
    k_self_max<<<gN, TPB, 0, stream>>>(f_asrc, f_adst, f_m, 1);
    k_edge_exp<<<gE, TPB, 0, stream>>>(ei, f_e, f_m, f_s, 1);
    k_self_sum<<<gN, TPB, 0, stream>>>(f_asrc, f_adst, f_m, f_s, 1);
    k_edge_agg<<<cdiv((long long)EE * OUT_DIM, TPB), TPB, 0, stream>>>(ei, f_e, f_s, f_h, f_agg, 1, OUT_DIM);
    k_self_agg<<<cdiv((long long)NN * OUT_DIM, TPB), TPB, 0, stream>>>(f_asrc, f_adst, f_m, f_s, f_h, f_agg, 1, OUT_DIM);
    k_fill_f32<<<cdiv(2 * HDIM, TPB), TPB, 0, stream>>>(f_sum, 0.f, 2 * HDIM);
    k_bn_stats<<<cdiv(NN, 1024), OUT_DIM, 0, stream>>>(f_agg, b2, f_sum, f_sq, OUT_DIM, 1024);
    k_bn_fin<<<1, OUT_DIM, 0, stream>>>(f_sum, f_sq, f_mu, f_rs, OUT_DIM);
    k_apply_out<<<cdiv((long long)NN * OUT_DIM, TPB), TPB, 0, stream>>>(f_agg, b2, f_mu, f_rs, g2, be2, out);
}